// Net_26774826123693
// MI455X (gfx1250) — compile-verified
//
#include <hip/hip_runtime.h>
#include <math.h>

#define B_GRAPHS 128
#define N_NODES  1024
#define E_EDGES  16384
#define F_INC    32
#define H_DIM    64
#define NCLS     10
#define K1N      820
#define K2N      656
#define K3N      525
#define NE       (B_GRAPHS * E_EDGES)   // 2,097,152
#define BN       (B_GRAPHS * N_NODES)   // 131,072

typedef float v2f __attribute__((ext_vector_type(2)));
typedef float v8f __attribute__((ext_vector_type(8)));

// ---------------------------------------------------------------------------
// Edge scatter: agg[dst] += x[src] * w   (memory/atomic bound part)
// ---------------------------------------------------------------------------
__global__ __launch_bounds__(256) void scatter_kernel(
    const float* __restrict__ x, const int* __restrict__ src,
    const int* __restrict__ dst, const float* __restrict__ w,
    float* __restrict__ agg, int ne, int C)
{
    const long long gid = (long long)blockIdx.x * blockDim.x + threadIdx.x;
    const int vecs = C >> 2;
    const int e = (int)(gid / vecs);
    if (e >= ne) return;
    const float wv = w[e];
    if (wv == 0.0f) return;
    const int c = (int)(gid - (long long)e * vecs) << 2;
    const int s = src[e], d = dst[e];
    const float4 xv = *(const float4*)(x + (size_t)s * C + c);
    float* o = agg + (size_t)d * C + c;
    atomicAdd(o + 0, xv.x * wv);
    atomicAdd(o + 1, xv.y * wv);
    atomicAdd(o + 2, xv.z * wv);
    atomicAdd(o + 3, xv.w * wv);
}

// ---------------------------------------------------------------------------
// Fused GraphConv node transform via f32 WMMA:
//   out = relu(agg @ Wr^T + x @ Wk^T + br), out dim H=64.
// One wave per 16-row tile; 4 v8f accumulators cover all 64 output columns.
// A tile (16x4 f32): lane L holds A[L%16][k0 + (L/16)*2 + v], v=0,1
// B tile (4x16 f32): lane L holds B[(L/16)*2 + v][L%16]  (B[k][n] = W[n][k])
// D (16x16 f32):     lane L, VGPR p -> D[p + (L/16)*8][L%16]
// ---------------------------------------------------------------------------
__global__ __launch_bounds__(256) void conv_mm_kernel(
    const float* __restrict__ x, const float* __restrict__ agg,
    const float* __restrict__ wr, const float* __restrict__ wk,
    const float* __restrict__ br, float* __restrict__ out, int Cin)
{
    const int lane = threadIdx.x & 31;
    const int wave = threadIdx.x >> 5;
    const int m0   = (blockIdx.x * 8 + wave) * 16;
    const int r    = lane & 15;          // row within tile (A) / row of W (B)
    const int kh   = (lane >> 4) << 1;   // k sub-offset for upper lane half

    const v8f vzero = {0.f, 0.f, 0.f, 0.f, 0.f, 0.f, 0.f, 0.f};
    v8f acc[4];
#pragma unroll
    for (int nt = 0; nt < 4; ++nt) acc[nt] = vzero;

    for (int k0 = 0; k0 < Cin; k0 += 4) {
        const v2f ax = *(const v2f*)(x   + (size_t)(m0 + r) * Cin + k0 + kh);
        const v2f ag = *(const v2f*)(agg + (size_t)(m0 + r) * Cin + k0 + kh);
#pragma unroll
        for (int nt = 0; nt < 4; ++nt) {
            const v2f bwr = *(const v2f*)(wr + (size_t)(nt * 16 + r) * Cin + k0 + kh);
            const v2f bwk = *(const v2f*)(wk + (size_t)(nt * 16 + r) * Cin + k0 + kh);
            acc[nt] = __builtin_amdgcn_wmma_f32_16x16x4_f32(
                false, ag, false, bwr, (short)0, acc[nt], false, false);
            acc[nt] = __builtin_amdgcn_wmma_f32_16x16x4_f32(
                false, ax, false, bwk, (short)0, acc[nt], false, false);
        }
    }

    const int rowoff = (lane >> 4) << 3;
#pragma unroll
    for (int nt = 0; nt < 4; ++nt) {
        const float bias = br[nt * 16 + r];
#pragma unroll
        for (int p = 0; p < 8; ++p) {
            float v = acc[nt][p] + bias;
            out[(size_t)(m0 + rowoff + p) * H_DIM + nt * 16 + r] = v > 0.f ? v : 0.f;
        }
    }
}

// ---------------------------------------------------------------------------
// TopK pooling: one 1024-thread block per graph. Bitonic sort (score desc,
// idx asc) in LDS; write gated features x*score compacted, and new_idx map.
// ---------------------------------------------------------------------------
__global__ __launch_bounds__(1024) void topk_pool_kernel(
    const float* __restrict__ h, const float* __restrict__ pw,
    float* __restrict__ xout, int* __restrict__ new_idx, int n, int k)
{
    __shared__ float s_score[1024];
    __shared__ int   s_idx[1024];
    __shared__ float s_pw[H_DIM];

    const int g = blockIdx.x;
    const int t = threadIdx.x;
    if (t < H_DIM) s_pw[t] = pw[t];
    __syncthreads();

    float nrm = 0.f;
#pragma unroll
    for (int j = 0; j < H_DIM; ++j) nrm += s_pw[j] * s_pw[j];
    nrm = sqrtf(nrm);

    float sc = -INFINITY;
    int   id = 0x7fffffff;
    if (t < n) {
        const float* hr = h + ((size_t)g * n + t) * H_DIM;
        float d = 0.f;
#pragma unroll
        for (int j = 0; j < H_DIM; ++j) d += hr[j] * s_pw[j];
        sc = tanhf(d / nrm);
        id = t;
    }
    s_score[t] = sc;
    s_idx[t]   = id;
    __syncthreads();

    // bitonic sort, 1024 elements, descending by score (idx asc tiebreak)
    for (int ksz = 2; ksz <= 1024; ksz <<= 1) {
        for (int j = ksz >> 1; j > 0; j >>= 1) {
            const int p = t ^ j;
            if (p > t) {
                const float a = s_score[t], b = s_score[p];
                const int ia = s_idx[t], ib = s_idx[p];
                const bool aFirst = (a > b) || (a == b && ia < ib); // desc order
                const bool descSeg = ((t & ksz) == 0);
                if (descSeg ? !aFirst : aFirst) {
                    s_score[t] = b; s_score[p] = a;
                    s_idx[t] = ib;  s_idx[p] = ia;
                }
            }
            __syncthreads();
        }
    }

    if (t < n) new_idx[(size_t)g * n + t] = -1;
    __syncthreads();

    if (t < k) {
        const int   old = s_idx[t];
        const float s   = s_score[t];
        new_idx[(size_t)g * n + old] = g * k + t;
        const float* hr = h + ((size_t)g * n + old) * H_DIM;
        float* xo = xout + ((size_t)g * k + t) * H_DIM;
#pragma unroll
        for (int j = 0; j < H_DIM; ++j) xo[j] = hr[j] * s;
    }
}

// ---------------------------------------------------------------------------
// Remap edge endpoints through new_idx; invalid edges -> (0,0,valid=0)
// ---------------------------------------------------------------------------
__global__ __launch_bounds__(256) void remap_kernel(
    const int* __restrict__ src_in, const int* __restrict__ dst_in,
    const float* __restrict__ valid_in, const int* __restrict__ new_idx,
    int* __restrict__ src_out, int* __restrict__ dst_out,
    float* __restrict__ valid_out, int ne)
{
    const int e = blockIdx.x * blockDim.x + threadIdx.x;
    if (e >= ne) return;
    const int s = src_in[e], d = dst_in[e];
    const float v = valid_in ? valid_in[e] : 1.0f;
    const int ns = new_idx[s], nd = new_idx[d];
    const bool ok = (v != 0.0f) && (ns >= 0) && (nd >= 0);
    src_out[e]   = ok ? ns : 0;
    dst_out[e]   = ok ? nd : 0;
    valid_out[e] = ok ? 1.0f : 0.0f;
}

// ---------------------------------------------------------------------------
// Readout: zsum[g, 0:64] += max over k nodes; zsum[g, 64:128] += mean
// ---------------------------------------------------------------------------
__global__ __launch_bounds__(64) void readout_kernel(
    const float* __restrict__ x, float* __restrict__ zsum, int k, int first)
{
    const int g = blockIdx.x;
    const int f = threadIdx.x;
    float mx = -INFINITY, sm = 0.f;
    for (int i = 0; i < k; ++i) {
        const float v = x[((size_t)g * k + i) * H_DIM + f];
        mx = fmaxf(mx, v);
        sm += v;
    }
    const float mean = sm / (float)k;
    if (first) {
        zsum[g * 128 + f]      = mx;
        zsum[g * 128 + 64 + f] = mean;
    } else {
        zsum[g * 128 + f]      += mx;
        zsum[g * 128 + 64 + f] += mean;
    }
}

// ---------------------------------------------------------------------------
// MLP head + log_softmax: one thread per graph (B=128, tiny)
// ---------------------------------------------------------------------------
__global__ __launch_bounds__(128) void mlp_kernel(
    const float* __restrict__ z,
    const float* __restrict__ w1, const float* __restrict__ b1,
    const float* __restrict__ w2, const float* __restrict__ b2,
    const float* __restrict__ w3, const float* __restrict__ b3,
    const float* __restrict__ w4, const float* __restrict__ b4,
    float* __restrict__ out)
{
    const int r = threadIdx.x;
    const float* zr = z + r * 128;
    float h1[64];
    for (int o = 0; o < 64; ++o) {
        float s = b1[o];
        for (int i = 0; i < 128; ++i) s += zr[i] * w1[o * 128 + i];
        h1[o] = s > 0.f ? s : 0.f;
    }
    float h2[32];
    for (int o = 0; o < 32; ++o) {
        float s = b2[o];
        for (int i = 0; i < 64; ++i) s += h1[i] * w2[o * 64 + i];
        h2[o] = s > 0.f ? s : 0.f;
    }
    float h3[16];
    for (int o = 0; o < 16; ++o) {
        float s = b3[o];
        for (int i = 0; i < 32; ++i) s += h2[i] * w3[o * 32 + i];
        h3[o] = s > 0.f ? s : 0.f;
    }
    float lg[NCLS];
    for (int o = 0; o < NCLS; ++o) {
        float s = b4[o];
        for (int i = 0; i < 16; ++i) s += h3[i] * w4[o * 16 + i];
        lg[o] = s;
    }
    float mx = lg[0];
    for (int c = 1; c < NCLS; ++c) mx = fmaxf(mx, lg[c]);
    float se = 0.f;
    for (int c = 0; c < NCLS; ++c) se += expf(lg[c] - mx);
    const float lse = mx + logf(se);
    for (int c = 0; c < NCLS; ++c) out[r * NCLS + c] = lg[c] - lse;
}

// ---------------------------------------------------------------------------
extern "C" void kernel_launch(void* const* d_in, const int* in_sizes, int n_in,
                              void* d_out, int out_size, void* d_ws, size_t ws_size,
                              hipStream_t stream)
{
    (void)in_sizes; (void)n_in; (void)out_size; (void)ws_size;
    const float* x      = (const float*)d_in[0];
    const int*   ei     = (const int*)d_in[1];
    const int*   src0   = ei;
    const int*   dst0   = ei + NE;
    const float* eattr  = (const float*)d_in[2];
    const float* c1_wr  = (const float*)d_in[3];
    const float* c1_br  = (const float*)d_in[4];
    const float* c1_wk  = (const float*)d_in[5];
    const float* p1_w   = (const float*)d_in[6];
    const float* c2_wr  = (const float*)d_in[7];
    const float* c2_br  = (const float*)d_in[8];
    const float* c2_wk  = (const float*)d_in[9];
    const float* p2_w   = (const float*)d_in[10];
    const float* c3_wr  = (const float*)d_in[11];
    const float* c3_br  = (const float*)d_in[12];
    const float* c3_wk  = (const float*)d_in[13];
    const float* p3_w   = (const float*)d_in[14];
    const float* l1_w   = (const float*)d_in[15];
    const float* l1_b   = (const float*)d_in[16];
    const float* l2_w   = (const float*)d_in[17];
    const float* l2_b   = (const float*)d_in[18];
    const float* l3_w   = (const float*)d_in[19];
    const float* l3_b   = (const float*)d_in[20];
    const float* l4_w   = (const float*)d_in[21];
    const float* l4_b   = (const float*)d_in[22];

    char* wsb = (char*)d_ws;
    size_t off = 0;
    auto carve = [&](size_t bytes) -> char* {
        char* p = wsb + off;
        off += (bytes + 255) & ~(size_t)255;
        return p;
    };
    float* fA     = (float*)carve((size_t)BN * H_DIM * 4);       // 33.5 MB
    float* fB     = (float*)carve((size_t)BN * H_DIM * 4);       // 33.5 MB
    float* agg    = (float*)carve((size_t)BN * H_DIM * 4);       // 33.5 MB
    int*   srcw   = (int*)carve((size_t)NE * 4);
    int*   dstw   = (int*)carve((size_t)NE * 4);
    float* validf = (float*)carve((size_t)NE * 4);
    int*   nidx   = (int*)carve((size_t)BN * 4);
    float* zsum   = (float*)carve((size_t)B_GRAPHS * 128 * 4);

    const int M1 = BN;                 // 131072 nodes after conv1
    const int M2 = B_GRAPHS * K1N;     // 104960
    const int M3 = B_GRAPHS * K2N;     // 83968

    // ---- conv1 (Cin=32) ----
    hipMemsetAsync(agg, 0, (size_t)M1 * F_INC * 4, stream);
    {
        const long long th = (long long)NE * (F_INC / 4);
        scatter_kernel<<<(unsigned)((th + 255) / 256), 256, 0, stream>>>(
            x, src0, dst0, eattr, agg, NE, F_INC);
    }
    conv_mm_kernel<<<M1 / 128, 256, 0, stream>>>(x, agg, c1_wr, c1_wk, c1_br, fA, F_INC);

    // ---- pool1 ----
    topk_pool_kernel<<<B_GRAPHS, 1024, 0, stream>>>(fA, p1_w, fB, nidx, N_NODES, K1N);
    remap_kernel<<<(NE + 255) / 256, 256, 0, stream>>>(
        src0, dst0, nullptr, nidx, srcw, dstw, validf, NE);
    readout_kernel<<<B_GRAPHS, 64, 0, stream>>>(fB, zsum, K1N, 1);

    // ---- conv2 (Cin=64, weight = valid) ----
    hipMemsetAsync(agg, 0, (size_t)M2 * H_DIM * 4, stream);
    {
        const long long th = (long long)NE * (H_DIM / 4);
        scatter_kernel<<<(unsigned)((th + 255) / 256), 256, 0, stream>>>(
            fB, srcw, dstw, validf, agg, NE, H_DIM);
    }
    conv_mm_kernel<<<M2 / 128, 256, 0, stream>>>(fB, agg, c2_wr, c2_wk, c2_br, fA, H_DIM);

    // ---- pool2 ----
    topk_pool_kernel<<<B_GRAPHS, 1024, 0, stream>>>(fA, p2_w, fB, nidx, K1N, K2N);
    remap_kernel<<<(NE + 255) / 256, 256, 0, stream>>>(
        srcw, dstw, validf, nidx, srcw, dstw, validf, NE);
    readout_kernel<<<B_GRAPHS, 64, 0, stream>>>(fB, zsum, K2N, 0);

    // ---- conv3 (Cin=64, weight = valid) ----
    hipMemsetAsync(agg, 0, (size_t)M3 * H_DIM * 4, stream);
    {
        const long long th = (long long)NE * (H_DIM / 4);
        scatter_kernel<<<(unsigned)((th + 255) / 256), 256, 0, stream>>>(
            fB, srcw, dstw, validf, agg, NE, H_DIM);
    }
    conv_mm_kernel<<<M3 / 128, 256, 0, stream>>>(fB, agg, c3_wr, c3_wk, c3_br, fA, H_DIM);

    // ---- pool3 (edges no longer needed) ----
    topk_pool_kernel<<<B_GRAPHS, 1024, 0, stream>>>(fA, p3_w, fB, nidx, K2N, K3N);
    readout_kernel<<<B_GRAPHS, 64, 0, stream>>>(fB, zsum, K3N, 0);

    // ---- MLP head ----
    mlp_kernel<<<1, 128, 0, stream>>>(zsum, l1_w, l1_b, l2_w, l2_b,
                                      l3_w, l3_b, l4_w, l4_b, (float*)d_out);
}